// ExpertChoiceRouter_4011499454964
// MI455X (gfx1250) — compile-verified
//
#include <hip/hip_runtime.h>
#include <hip/hip_bf16.h>
#include <math.h>

// ---------------------------------------------------------------------------
// Expert-Choice Router for MI455X (gfx1250, wave32)
//   tokens T = 4*4096 = 16384, d_model D = 2048, experts E = 16, top_k = 1024
//   Gate GEMM is HBM-bound (134 MB of x @ 23.3 TB/s ~= 5.8 us floor); done
//   with V_WMMA_F32_16X16X4_F32 so fp32 precision is preserved for the
//   rank-sensitive top-k. Split-K (x4) raises in-flight bytes to cover the
//   HBM bandwidth-delay product; partials reduced in fixed order for
//   graph-replay determinism.
// ---------------------------------------------------------------------------

typedef float v2f __attribute__((ext_vector_type(2)));
typedef float v8f __attribute__((ext_vector_type(8)));

static constexpr int T_TOKENS = 16384;   // 4 * 4096
static constexpr int D_MODEL  = 2048;
static constexpr int N_EXP    = 16;
static constexpr int TOPK     = 1024;    // T / E * capacity_factor

// --------------------------------------------------------------------------
// K1: gate GEMM.  One wave computes a 16-token x 16-expert logit tile over a
//     K-chunk of klen.  A (16x4 fp32): lanes 0-15 = row M with K{0,1},
//     lanes 16-31 = K{2,3}; B mirrors with N = lane&15.  So lane l and lane
//     l+16 together read 16 contiguous bytes of one row per WMMA step.
//     Partials stored transposed: Pt[chunk][e][t].
// --------------------------------------------------------------------------
__global__ void gate_gemm_wmma(const float* __restrict__ x,
                               const float* __restrict__ W,
                               float* __restrict__ Pt,
                               int klen) {
    const int lane  = threadIdx.x & 31;
    const int wave  = threadIdx.x >> 5;
    const int tile  = blockIdx.x * (blockDim.x >> 5) + wave;  // 16-token tile
    const int kc    = blockIdx.y;                              // K chunk
    const int row   = lane & 15;   // token-in-tile (A) / expert (B)
    const int khalf = lane >> 4;   // 0 -> K{0,1}, 1 -> K{2,3}
    const int k0    = kc * klen;

    const float* __restrict__ xrow =
        x + (size_t)(tile * 16 + row) * D_MODEL + k0 + khalf * 2;
    const float* __restrict__ wrow =
        W + (size_t)row * D_MODEL + k0 + khalf * 2;

    v8f c = {};
    for (int k = 0; k < klen; k += 64) {
        // one global_prefetch_b8 per 256B of row per lane (2KB ahead)
        __builtin_prefetch(xrow + k + 512, 0, 1);
#pragma unroll
        for (int kk = 0; kk < 64; kk += 4) {
            v2f a = *(const v2f*)(xrow + k + kk);
            v2f b = *(const v2f*)(wrow + k + kk);
            c = __builtin_amdgcn_wmma_f32_16x16x4_f32(false, a, false, b,
                                                      (short)0, c, false, false);
        }
    }

    // C/D layout: lane l -> N = l&15 (expert), VGPR i -> M = i + 8*(l>>4)
    const int e         = lane & 15;
    const int tokenBase = tile * 16 + khalf * 8;
    float* __restrict__ dst = Pt + (size_t)kc * N_EXP * T_TOKENS;
#pragma unroll
    for (int i = 0; i < 8; ++i) {
        dst[(size_t)e * T_TOKENS + tokenBase + i] = c[i];
    }
}

// --------------------------------------------------------------------------
// K1b: fixed-order reduction of split-K partials (deterministic fp32 sum).
// --------------------------------------------------------------------------
__global__ void reduce_partials(const float* __restrict__ Pt,
                                float* __restrict__ Lt,
                                int chunks) {
    const int i = blockIdx.x * blockDim.x + threadIdx.x;
    const size_t S = (size_t)N_EXP * T_TOKENS;
    if (i < (int)S) {
        float s = Pt[i];
        for (int c = 1; c < chunks; ++c) s += Pt[(size_t)c * S + i];
        Lt[i] = s;
    }
}

// --------------------------------------------------------------------------
// K0: zero expert_of / selected (workspace is poisoned; must be deterministic)
// --------------------------------------------------------------------------
__global__ void init_flags(int* __restrict__ expert_of, int* __restrict__ selected) {
    int t = blockIdx.x * blockDim.x + threadIdx.x;
    if (t < T_TOKENS) { expert_of[t] = 0; selected[t] = 0; }
}

__device__ __forceinline__ unsigned sortable_key(float f) {
    unsigned u = __float_as_uint(f);
    return (u & 0x80000000u) ? ~u : (u | 0x80000000u);  // order-preserving map
}

// --------------------------------------------------------------------------
// K2: per-expert exact top-k threshold via MSB->LSB byte radix select.
//     thresh[e] = key of the 1024th-largest logit, remain[e] = #ties to accept.
// --------------------------------------------------------------------------
__global__ void radix_threshold(const float* __restrict__ Lt,
                                unsigned* __restrict__ thresh,
                                int* __restrict__ remain) {
    const int e = blockIdx.x;
    __shared__ unsigned hist[256];
    __shared__ unsigned sel_bin;
    __shared__ int      new_k;

    unsigned prefix = 0, mask = 0;
    int k = TOPK;

    for (int pass = 3; pass >= 0; --pass) {
        const int shift = pass * 8;
        for (int i = threadIdx.x; i < 256; i += blockDim.x) hist[i] = 0;
        __syncthreads();
        for (int t = threadIdx.x; t < T_TOKENS; t += blockDim.x) {
            unsigned key = sortable_key(Lt[(size_t)e * T_TOKENS + t]);
            if ((key & mask) == prefix)
                atomicAdd(&hist[(key >> shift) & 0xFFu], 1u);
        }
        __syncthreads();
        if (threadIdx.x == 0) {
            int kk = k;
            for (int b = 255; b >= 0; --b) {
                int c = (int)hist[b];
                if (kk <= c) { sel_bin = (unsigned)b; new_k = kk; break; }
                kk -= c;
            }
        }
        __syncthreads();
        prefix |= (sel_bin << shift);
        mask   |= (0xFFu << shift);
        k = new_k;
        __syncthreads();
    }
    if (threadIdx.x == 0) { thresh[e] = prefix; remain[e] = k; }
}

// --------------------------------------------------------------------------
// K3: per-expert selection. Ties at the threshold are accepted in increasing
//     token-index order (matches jax top_k tie-breaking) via a block-wide
//     exclusive scan. Last-wins == max expert id -> atomicMax.
// --------------------------------------------------------------------------
__global__ void select_scan(const float* __restrict__ Lt,
                            const unsigned* __restrict__ thresh,
                            const int* __restrict__ remain,
                            int* __restrict__ expert_of,
                            int* __restrict__ selected) {
    const int e = blockIdx.x;
    const unsigned Tkey = thresh[e];
    const int r = remain[e];
    const int BT = blockDim.x;   // 256

    __shared__ int scan[256];
    __shared__ int base;
    if (threadIdx.x == 0) base = 0;
    __syncthreads();

    for (int start = 0; start < T_TOKENS; start += BT) {
        const int t = start + threadIdx.x;
        unsigned key = sortable_key(Lt[(size_t)e * T_TOKENS + t]);
        const int eq = (key == Tkey) ? 1 : 0;
        const int gt = (key > Tkey) ? 1 : 0;

        scan[threadIdx.x] = eq;
        __syncthreads();
        for (int off = 1; off < BT; off <<= 1) {         // Hillis-Steele
            int v = (threadIdx.x >= off) ? scan[threadIdx.x - off] : 0;
            __syncthreads();
            scan[threadIdx.x] += v;
            __syncthreads();
        }
        const int incl = scan[threadIdx.x];
        const int excl = incl - eq;

        if (gt || (eq && (base + excl) < r)) {
            atomicMax(&expert_of[t], e);
            selected[t] = 1;
        }
        __syncthreads();
        if (threadIdx.x == BT - 1) base += incl;
        __syncthreads();
    }
}

// --------------------------------------------------------------------------
// K4: softmax over 16 experts per token; emit [indices | weights | aux_loss]
// --------------------------------------------------------------------------
__global__ void finalize(const float* __restrict__ Lt,
                         const int* __restrict__ expert_of,
                         const int* __restrict__ selected,
                         float* __restrict__ out) {
    const int t = blockIdx.x * blockDim.x + threadIdx.x;
    if (t >= T_TOKENS) return;

    float l[N_EXP];
    float m = -INFINITY;
#pragma unroll
    for (int e = 0; e < N_EXP; ++e) {
        l[e] = Lt[(size_t)e * T_TOKENS + t];
        m = fmaxf(m, l[e]);
    }
    float s = 0.f;
#pragma unroll
    for (int e = 0; e < N_EXP; ++e) s += expf(l[e] - m);

    const int eo  = expert_of[t];
    const float p = expf(l[eo] - m) / s;
    const float w = selected[t] ? p : 1.0f;

    out[t]            = (float)eo;   // expert_indices
    out[T_TOKENS + t] = w;           // expert_weights
    if (t == 0) out[2 * T_TOKENS] = 0.0f;   // aux_loss
}

// --------------------------------------------------------------------------
extern "C" void kernel_launch(void* const* d_in, const int* in_sizes, int n_in,
                              void* d_out, int out_size, void* d_ws, size_t ws_size,
                              hipStream_t stream) {
    const float* x = (const float*)d_in[0];   // [4,4096,2048] fp32
    const float* W = (const float*)d_in[1];   // [16,2048]     fp32
    float* out = (float*)d_out;

    const size_t Sbytes = (size_t)N_EXP * T_TOKENS * sizeof(float);  // 1 MB

    // Choose split-K factor based on available workspace (deterministic for a
    // given ws_size). chunks=4 gives ~4096 waves -> covers HBM BDP.
    int chunks = 4;
    const size_t flagsBytes = (size_t)2 * T_TOKENS * sizeof(int) + 4096;
    if (ws_size < (size_t)(chunks + 1) * Sbytes + flagsBytes) chunks = 1;

    char* ws = (char*)d_ws;
    float*    Lt        = (float*)ws;                              // [E][T] logits
    float*    Pt        = (chunks == 1) ? Lt : (float*)(ws + Sbytes);
    char*     tail      = ws + (size_t)(chunks == 1 ? 1 : chunks + 1) * Sbytes;
    unsigned* thresh    = (unsigned*)tail;
    int*      remain    = (int*)(tail + N_EXP * 4);
    int*      expert_of = (int*)(tail + 2 * N_EXP * 4);            // 64 KB
    int*      selected  = expert_of + T_TOKENS;                    // 64 KB

    // K1: gate GEMM (WMMA). 1024 tiles x chunks K-chunks, 8 waves/block.
    gate_gemm_wmma<<<dim3(T_TOKENS / 16 / 8, chunks), dim3(256), 0, stream>>>(
        x, W, Pt, D_MODEL / chunks);

    // K1b: deterministic fixed-order reduction of partials
    if (chunks > 1) {
        reduce_partials<<<dim3((N_EXP * T_TOKENS + 255) / 256), dim3(256), 0,
                          stream>>>(Pt, Lt, chunks);
    }

    // K0: init flags
    init_flags<<<dim3((T_TOKENS + 255) / 256), dim3(256), 0, stream>>>(expert_of,
                                                                       selected);

    // K2: per-expert exact top-k threshold
    radix_threshold<<<dim3(N_EXP), dim3(256), 0, stream>>>(Lt, thresh, remain);

    // K3: per-expert selection with index-ordered tie-breaking
    select_scan<<<dim3(N_EXP), dim3(256), 0, stream>>>(Lt, thresh, remain,
                                                       expert_of, selected);

    // K4: softmax + output assembly
    finalize<<<dim3((T_TOKENS + 255) / 256), dim3(256), 0, stream>>>(Lt, expert_of,
                                                                     selected, out);
}